// GNCELoss_24000277250552
// MI455X (gfx1250) — compile-verified
//
#include <hip/hip_runtime.h>

typedef __attribute__((ext_vector_type(2))) float v2f;
typedef __attribute__((ext_vector_type(4))) float v4f;
typedef __attribute__((ext_vector_type(8))) float v8f;

constexpr int B = 32, T = 1024, D = 768, KNEG = 16;
constexpr int WAVE = 32;
constexpr int NV4 = D / (WAVE * 4);    // 6 float4 chunks per lane per row
constexpr float TAU_INV = 10.0f;       // 1/0.1
constexpr int ROWSTR = D + 4;          // 772: bank-conflict-free LDS row stride (div by 4)
constexpr int GSTR = 33;               // padded gram stride

__device__ __forceinline__ float wave_sum(float v) {
#pragma unroll
  for (int m = 16; m > 0; m >>= 1) v += __shfl_xor(v, m, 32);
  return v;
}

__device__ __forceinline__ float log_sigmoid(float x) {
  // numerically stable: min(x,0) - log1p(exp(-|x|))
  return fminf(x, 0.0f) - log1pf(expf(-fabsf(x)));
}

__global__ __launch_bounds__(512) void gnce_pos_kernel(
    const float* __restrict__ inter, const int* __restrict__ targets,
    const float* __restrict__ wte, const float* __restrict__ gamma,
    const float* __restrict__ beta, int shift, float* __restrict__ acc) {
  extern __shared__ float smem[];
  float* sG  = smem;                   // D        gamma
  float* sBt = sG + D;                 // D        beta
  float* sHn = sBt + D;                // B*ROWSTR normalized hidden rows
  float* sTn = sHn + B * ROWSTR;       // B*ROWSTR normalized target rows
  float* sGm = sTn + B * ROWSTR;       // B*GSTR   gram matrix

  const int t  = blockIdx.x;           // time position
  const int tid = threadIdx.x;
  const int wv = tid >> 5;             // wave id (0..15)
  const int ln = tid & 31;             // lane id

  for (int d = tid; d < D; d += blockDim.x) { sG[d] = gamma[d]; sBt[d] = beta[d]; }
  __syncthreads();

  const v4f* sG4  = (const v4f*)sG;
  const v4f* sBt4 = (const v4f*)sBt;

  // ---- Phase A: normalize 64 rows (one wave per row, 4 rows per wave) ----
  // Lane l owns float4 chunks {j*32 + l, j=0..5}: 512B contiguous per wave per j.
  for (int rr = 0; rr < 4; ++rr) {
    const int row = wv * 4 + rr;       // 0..63
    const int b = row & (B - 1);
    v4f v[NV4];
    if (row < B) {
      // hidden row: LayerNorm then L2 normalize; streamed once -> non-temporal
      const v4f* x = (const v4f*)(inter + ((size_t)b * T + t) * D);
      float s = 0.f;
#pragma unroll
      for (int j = 0; j < NV4; ++j) {
        v[j] = __builtin_nontemporal_load(x + (j * WAVE + ln));
        s += v[j][0] + v[j][1] + v[j][2] + v[j][3];
      }
      s = wave_sum(s);
      const float mu = s * (1.0f / D);
      float s2 = 0.f;
#pragma unroll
      for (int j = 0; j < NV4; ++j) {
#pragma unroll
        for (int c = 0; c < 4; ++c) { const float dd = v[j][c] - mu; s2 += dd * dd; }
      }
      s2 = wave_sum(s2);
      const float rstd = rsqrtf(s2 * (1.0f / D) + 1e-5f);
      float n2 = 0.f;
#pragma unroll
      for (int j = 0; j < NV4; ++j) {
        const v4f g = sG4[j * WAVE + ln];
        const v4f bt = sBt4[j * WAVE + ln];
        const v4f y = (v[j] - mu) * rstd * g + bt;
        v[j] = y;
        n2 += y[0] * y[0] + y[1] * y[1] + y[2] * y[2] + y[3] * y[3];
      }
      n2 = wave_sum(n2);
      const float inv = 1.0f / fmaxf(sqrtf(n2), 1e-12f);
      v4f* dst = (v4f*)(sHn + b * ROWSTR);
#pragma unroll
      for (int j = 0; j < NV4; ++j) dst[j * WAVE + ln] = v[j] * inv;
    } else {
      // target embedding row: gather + L2 normalize; keep regular-temporal
      // (wte table fits 192MB L2 and is reused by the second loss term)
      const int idx = targets[b * T + t + shift];
      const v4f* x = (const v4f*)(wte + (size_t)idx * D);
      float n2 = 0.f;
#pragma unroll
      for (int j = 0; j < NV4; ++j) {
        v[j] = x[j * WAVE + ln];
        n2 += v[j][0] * v[j][0] + v[j][1] * v[j][1] + v[j][2] * v[j][2] + v[j][3] * v[j][3];
      }
      n2 = wave_sum(n2);
      const float inv = 1.0f / fmaxf(sqrtf(n2), 1e-12f);
      v4f* dst = (v4f*)(sTn + b * ROWSTR);
#pragma unroll
      for (int j = 0; j < NV4; ++j) dst[j * WAVE + ln] = v[j] * inv;
    }
  }
  __syncthreads();

  // ---- Phase B: G = Hn * Tn^T via V_WMMA_F32_16X16X4_F32, 4 tiles, 1 wave each ----
  if (wv < 4) {
    const int tm = wv & 1, tn = wv >> 1;
    const int l = ln & 15, hi = ln >> 4;
    // f32 A 16x4 layout: lane l holds {A[l][2*hi], A[l][2*hi+1]}; B mirrors with
    // lane l holding column l of B, i.e. a row of Tn -> both operands are rows.
    const float* hb = sHn + (tm * 16 + l) * ROWSTR + 2 * hi;
    const float* tb = sTn + (tn * 16 + l) * ROWSTR + 2 * hi;
    v8f c = {0.f, 0.f, 0.f, 0.f, 0.f, 0.f, 0.f, 0.f};
#pragma unroll 8
    for (int k0 = 0; k0 < D; k0 += 4) {
      v2f a = *(const v2f*)(hb + k0);
      v2f bm = *(const v2f*)(tb + k0);
      c = __builtin_amdgcn_wmma_f32_16x16x4_f32(
          /*neg_a=*/false, a, /*neg_b=*/false, bm,
          /*c_mod=*/(short)0, c, /*reuse_a=*/false, /*reuse_b=*/false);
    }
    // C/D layout: VGPR r -> lanes 0-15: (M=r, N=l); lanes 16-31: (M=r+8, N=l)
#pragma unroll
    for (int r = 0; r < 8; ++r) {
      const int m = tm * 16 + hi * 8 + r;
      const int n = tn * 16 + l;
      sGm[m * GSTR + n] = c[r];
    }
  }
  __syncthreads();

  // ---- Phase C: banded log-sigmoid terms, wave reduction, one atomic ----
  if (tid < 32) {
    const int b = tid;
    const float pos = sGm[b * GSTR + b] * TAU_INV;
    float term = log_sigmoid(pos);
    float nsum = 0.f;
#pragma unroll
    for (int k = 1; k <= KNEG; ++k) {
      const int cix = (b - k) & (B - 1);
      nsum += log_sigmoid(-sGm[b * GSTR + cix] * TAU_INV);
    }
    term += nsum * (1.0f / KNEG);
    const float tot = wave_sum(term);
    if (ln == 0) atomicAdd(acc, tot);
  }
}

__global__ void gnce_init(float* acc) {
  if (threadIdx.x < 2) acc[threadIdx.x] = 0.f;
}

__global__ void gnce_final(const float* __restrict__ ce_in,
                           const float* __restrict__ acc,
                           float* __restrict__ out) {
  const float nce2 = -acc[0] / (float)(B * (T - 2));
  const float nce3 = -acc[1] / (float)(B * (T - 3));
  const float ncec = 0.5f * nce2 + 0.25f * nce3;
  const float ce = ce_in[0];
  const float ratio = ce / fmaxf(ncec, 1e-6f);
  const float alpha = fminf(fmaxf(0.3f * ratio, 0.01f), 10.0f);
  out[0] = ce + alpha * ncec;
}

extern "C" void kernel_launch(void* const* d_in, const int* in_sizes, int n_in,
                              void* d_out, int out_size, void* d_ws, size_t ws_size,
                              hipStream_t stream) {
  const float* ce     = (const float*)d_in[0];
  const float* inter9 = (const float*)d_in[1];
  const float* inter8 = (const float*)d_in[2];
  const int*   tgts   = (const int*)d_in[3];
  const float* wte    = (const float*)d_in[4];
  const float* g2     = (const float*)d_in[5];
  const float* b2     = (const float*)d_in[6];
  const float* g3     = (const float*)d_in[7];
  const float* b3     = (const float*)d_in[8];
  float* acc = (float*)d_ws;      // acc[0] = sum for nce2, acc[1] = sum for nce3
  float* out = (float*)d_out;

  const size_t lds = (size_t)(2 * D + 2 * B * ROWSTR + B * GSTR) * sizeof(float);

  hipLaunchKernelGGL(gnce_init, dim3(1), dim3(32), 0, stream, acc);
  hipLaunchKernelGGL(gnce_pos_kernel, dim3(T - 2), dim3(512), lds, stream,
                     inter9, tgts, wte, g2, b2, 2, acc + 0);
  hipLaunchKernelGGL(gnce_pos_kernel, dim3(T - 3), dim3(512), lds, stream,
                     inter8, tgts, wte, g3, b3, 3, acc + 1);
  hipLaunchKernelGGL(gnce_final, dim3(1), dim3(1), 0, stream, ce, acc, out);
}